// SAGE_87101936762934
// MI455X (gfx1250) — compile-verified
//
#include <hip/hip_runtime.h>
#include <hip/hip_bf16.h>

typedef float v2f __attribute__((ext_vector_type(2)));
typedef float v8f __attribute__((ext_vector_type(8)));

// ---------------------------------------------------------------------------
// Zero a buffer (float4 granularity)
// ---------------------------------------------------------------------------
__global__ __launch_bounds__(256)
void sage_zero_kernel(float4* __restrict__ p, long n4) {
    long i = (long)blockIdx.x * 256 + threadIdx.x;
    if (i < n4) p[i] = make_float4(0.f, 0.f, 0.f, 0.f);
}

// ---------------------------------------------------------------------------
// Degree count (float atomic increments; exact up to 2^24)
// ---------------------------------------------------------------------------
__global__ __launch_bounds__(256)
void sage_deg_count_kernel(const int* __restrict__ dst, float* __restrict__ deg, int E) {
    int e = blockIdx.x * 256 + threadIdx.x;
    if (e < E) unsafeAtomicAdd(&deg[dst[e]], 1.0f);
}

__global__ __launch_bounds__(256)
void sage_deg_inv_kernel(float* __restrict__ deg, int n) {
    int i = blockIdx.x * 256 + threadIdx.x;
    if (i < n) deg[i] = 1.0f / fmaxf(deg[i], 1.0f);
}

// ---------------------------------------------------------------------------
// Edge scatter-add: one wave per edge, lane owns a float4 chunk of the row.
// Gather of x[src] rows is fully coalesced (512B per wave) and L2-resident
// (node table = 51.2 MB < 192 MB L2). Scatter uses native f32 atomics
// (no return -> STOREcnt path, resolved at L2 atomic units).
// ---------------------------------------------------------------------------
__global__ __launch_bounds__(256)
void sage_scatter_kernel(const float* __restrict__ x, const int* __restrict__ src,
                         const int* __restrict__ dst, float* __restrict__ agg, int E) {
    long t = (long)blockIdx.x * 256 + threadIdx.x;
    long e = t >> 5;
    if (e >= E) return;
    int c = ((int)t & 31) * 4;
    int s = src[e];
    int d = dst[e];
    float4 v = *(const float4*)(x + (long)s * 128 + c);
    float* p = agg + (long)d * 128 + c;
    unsafeAtomicAdd(p + 0, v.x);
    unsafeAtomicAdd(p + 1, v.y);
    unsafeAtomicAdd(p + 2, v.z);
    unsafeAtomicAdd(p + 3, v.w);
}

// ---------------------------------------------------------------------------
// Fused SAGE layer GEMM:
//   out[r, :] = act( (agg[r,:] * dinv[r]) @ Wl + bl + xin[r,:] @ Wr )
// K = 128 fixed. One wave computes a 16-row M-tile across all NOUT columns
// using V_WMMA_F32_16X16X4_F32 (fp32, exact). Weights staged in LDS in
// 32-row k-panels stored K-PAIR INTERLEAVED: element (k,c) lives at
// (k/2)*LDP + 2c + (k&1), so each B fragment {(k,c),(k+1,c)} is one
// contiguous 8B ds_load_b64 directly into an even VGPR pair (no repacking).
// LDP = 2*NOUT + 32  =>  LDP mod 64 == 32 banks, so the two 16-lane halves
// (k vs k+2) hit disjoint 32-bank sets: conflict-free.
// ---------------------------------------------------------------------------
template<int NOUT, bool RELU>
__global__ __launch_bounds__(256)
void sage_gemm_kernel(const float* __restrict__ agg, const float* __restrict__ dinv,
                      const float* __restrict__ xin,
                      const float* __restrict__ Wl, const float* __restrict__ bl,
                      const float* __restrict__ Wr,
                      float* __restrict__ out, int n) {
    constexpr int K   = 128;
    constexpr int KP  = 32;              // k-panel rows staged in LDS
    constexpr int LDP = 2 * NOUT + 32;   // floats per k-PAIR row (mod 64 == 32)
    constexpr int NT  = NOUT / 16;       // N-tiles per wave

    __shared__ float sWl[(KP / 2) * LDP];
    __shared__ float sWr[(KP / 2) * LDP];
    __shared__ float sbias[NOUT];

    const int lane = threadIdx.x & 31;
    const int wave = threadIdx.x >> 5;

    for (int i = threadIdx.x; i < NOUT; i += 256) sbias[i] = bl[i];

    // This wave's 16-row M tile; clamp tail (duplicate waves write identical data)
    long tile = (long)blockIdx.x * 8 + wave;
    long row0 = tile * 16;
    if (row0 > (long)n - 16) row0 = (long)n - 16;

    const int m  = lane & 15;            // A-fragment row within tile
    const int kg = (lane >> 4) * 2;      // A/B fragment k sub-offset (0 or 2)
    const float scale = dinv[row0 + m];  // fold mean (1/deg) into A load

    v8f acc[NT] = {};

    const float* aL = agg + (row0 + m) * K;
    const float* aR = xin + (row0 + m) * K;

    for (int kp = 0; kp < K; kp += KP) {
        __syncthreads();
        // Cooperative panel load: float4 from two adjacent k-rows, written as
        // interleaved {row k, row k+1} pairs (8B ds stores).
        for (int i = threadIdx.x; i < (KP / 2) * (NOUT / 4); i += 256) {
            int r2 = i / (NOUT / 4);           // pair-of-rows index
            int c  = (i % (NOUT / 4)) * 4;
            const float* g0 = Wl + (long)(kp + 2 * r2) * NOUT + c;
            float4 a0 = *(const float4*)g0;
            float4 a1 = *(const float4*)(g0 + NOUT);
            float* s = sWl + r2 * LDP + 2 * c;
            ((v2f*)s)[0] = (v2f){a0.x, a1.x};
            ((v2f*)s)[1] = (v2f){a0.y, a1.y};
            ((v2f*)s)[2] = (v2f){a0.z, a1.z};
            ((v2f*)s)[3] = (v2f){a0.w, a1.w};
            const float* h0 = Wr + (long)(kp + 2 * r2) * NOUT + c;
            float4 b0 = *(const float4*)h0;
            float4 b1 = *(const float4*)(h0 + NOUT);
            float* q = sWr + r2 * LDP + 2 * c;
            ((v2f*)q)[0] = (v2f){b0.x, b1.x};
            ((v2f*)q)[1] = (v2f){b0.y, b1.y};
            ((v2f*)q)[2] = (v2f){b0.z, b1.z};
            ((v2f*)q)[3] = (v2f){b0.w, b1.w};
        }
        __syncthreads();

#pragma unroll
        for (int kk = 0; kk < KP; kk += 4) {
            const int k = kp + kk;
            v2f a_l = *(const v2f*)(aL + k + kg);   // A 16x4 f32 fragment (2 VGPRs)
            v2f a_r = *(const v2f*)(aR + k + kg);
            a_l *= scale;                           // mean aggregation scale
            const int prow = (kk + kg) >> 1;        // k-pair row in LDS
            const float* pbl = sWl + prow * LDP;
            const float* pbr = sWr + prow * LDP;
#pragma unroll
            for (int t = 0; t < NT; ++t) {
                const int col = t * 16 + (lane & 15);
                v2f b_l = *(const v2f*)(pbl + 2 * col);   // one ds_load_b64
                v2f b_r = *(const v2f*)(pbr + 2 * col);
                acc[t] = __builtin_amdgcn_wmma_f32_16x16x4_f32(
                    false, a_l, false, b_l, (short)0, acc[t], false, false);
                acc[t] = __builtin_amdgcn_wmma_f32_16x16x4_f32(
                    false, a_r, false, b_r, (short)0, acc[t], false, false);
            }
        }
    }

    // C/D layout: VGPR v holds (M = v + 8*(lane>>4), N = lane&15)
    const int mg = (lane >> 4) * 8;
#pragma unroll
    for (int t = 0; t < NT; ++t) {
        const int col = t * 16 + (lane & 15);
        const float b = sbias[col];
#pragma unroll
        for (int v = 0; v < 8; ++v) {
            float val = acc[t][v] + b;
            if (RELU) val = fmaxf(val, 0.f);
            out[(row0 + mg + v) * NOUT + col] = val;
        }
    }
}

// ---------------------------------------------------------------------------
// Row-wise L2 normalize (D=64): one wave per row, shuffle reduction
// ---------------------------------------------------------------------------
__global__ __launch_bounds__(256)
void sage_l2norm_kernel(float* __restrict__ out, int n) {
    long row = (long)blockIdx.x * 8 + (threadIdx.x >> 5);
    if (row >= n) return;
    int lane = threadIdx.x & 31;
    float* p = out + row * 64 + lane * 2;
    float2 v = *(float2*)p;
    float s = v.x * v.x + v.y * v.y;
#pragma unroll
    for (int off = 16; off > 0; off >>= 1) s += __shfl_xor(s, off, 32);
    float inv = 1.0f / fmaxf(sqrtf(s), 1e-12f);
    v.x *= inv; v.y *= inv;
    *(float2*)p = v;
}

// ---------------------------------------------------------------------------
extern "C" void kernel_launch(void* const* d_in, const int* in_sizes, int n_in,
                              void* d_out, int out_size, void* d_ws, size_t ws_size,
                              hipStream_t stream) {
    const float* x   = (const float*)d_in[0];
    const int*   ei  = (const int*)d_in[1];
    const float* Wl1 = (const float*)d_in[2];
    const float* bl1 = (const float*)d_in[3];
    const float* Wr1 = (const float*)d_in[4];
    const float* Wl2 = (const float*)d_in[5];
    const float* bl2 = (const float*)d_in[6];
    const float* Wr2 = (const float*)d_in[7];
    const float* Wl3 = (const float*)d_in[8];
    const float* bl3 = (const float*)d_in[9];
    const float* Wr3 = (const float*)d_in[10];

    const int n = in_sizes[0] / 128;
    const int E = in_sizes[1] / 2;
    const int* src = ei;
    const int* dst = ei + E;

    // Workspace layout
    float* deg = (float*)d_ws;                          // [n] -> becomes 1/max(deg,1)
    size_t npad = ((size_t)n + 255) & ~(size_t)255;
    float* agg = deg + npad;                            // [n,128]
    float* h1  = agg + (size_t)n * 128;                 // [n,128]
    float* h2  = h1 + (size_t)n * 128;                  // [n,128]
    float* out = (float*)d_out;                         // [n,64]

    auto zero = [&](float* p, size_t cnt) {
        long n4 = (long)(cnt / 4);
        sage_zero_kernel<<<dim3((n4 + 255) / 256), dim3(256), 0, stream>>>((float4*)p, n4);
    };

    // Degrees (shared across all 3 layers)
    zero(deg, npad);
    sage_deg_count_kernel<<<dim3((E + 255) / 256), dim3(256), 0, stream>>>(dst, deg, E);
    sage_deg_inv_kernel<<<dim3((n + 255) / 256), dim3(256), 0, stream>>>(deg, n);

    const int sblocks = (int)(((long)E * 32 + 255) / 256);
    const int gblocks = (int)((((long)n + 15) / 16 + 7) / 8);

    // Layer 1: h1 = relu(mean @ Wl1 + bl1 + x @ Wr1)
    zero(agg, (size_t)n * 128);
    sage_scatter_kernel<<<dim3(sblocks), dim3(256), 0, stream>>>(x, src, dst, agg, E);
    sage_gemm_kernel<128, true><<<dim3(gblocks), dim3(256), 0, stream>>>(
        agg, deg, x, Wl1, bl1, Wr1, h1, n);

    // Layer 2: h2 = relu(mean @ Wl2 + bl2 + h1 @ Wr2)
    zero(agg, (size_t)n * 128);
    sage_scatter_kernel<<<dim3(sblocks), dim3(256), 0, stream>>>(h1, src, dst, agg, E);
    sage_gemm_kernel<128, true><<<dim3(gblocks), dim3(256), 0, stream>>>(
        agg, deg, h1, Wl2, bl2, Wr2, h2, n);

    // Layer 3: out = mean @ Wl3 + bl3 + h2 @ Wr3, then row L2 normalize
    zero(agg, (size_t)n * 128);
    sage_scatter_kernel<<<dim3(sblocks), dim3(256), 0, stream>>>(h2, src, dst, agg, E);
    sage_gemm_kernel<64, false><<<dim3(gblocks), dim3(256), 0, stream>>>(
        agg, deg, h2, Wl3, bl3, Wr3, out, n);

    sage_l2norm_kernel<<<dim3((int)(((long)n + 7) / 8)), dim3(256), 0, stream>>>(out, n);
}